// LinkLevelModel_83683142795431
// MI455X (gfx1250) — compile-verified
//
#include <hip/hip_runtime.h>
#include <math.h>

typedef float v2f __attribute__((ext_vector_type(2)));
typedef float v8f __attribute__((ext_vector_type(8)));

#define LRELU_SLOPE 0.2f

// ---------------- helpers: order-preserving float <-> uint for atomicMax ----
__device__ __forceinline__ unsigned fenc(float f) {
  unsigned b = __float_as_uint(f);
  return b ^ ((unsigned)((int)b >> 31) | 0x80000000u);
}
__device__ __forceinline__ float fdec(unsigned u) {
  unsigned b = (u & 0x80000000u) ? (u ^ 0x80000000u) : ~u;
  return __uint_as_float(b);
}

// ---------------- GEMM: C[n,256] = X[n,K] @ W[K,256], f32 WMMA 16x16x4 ------
// One wave computes a 16(M) x 64(N) stripe: 4 accumulators, A reused 4x.
__global__ __launch_bounds__(256) void gemm_f32_wmma(
    const float* __restrict__ X, const float* __restrict__ Wm,
    float* __restrict__ C, int n, int K) {
  const int wave  = (int)((blockIdx.x * blockDim.x + threadIdx.x) >> 5);
  const int lane  = (int)(threadIdx.x & 31);
  const int mTile = wave >> 2;     // 256/64 = 4 N-stripes per M row-tile
  const int nTile = wave & 3;
  const int row0  = mTile * 16;
  if (row0 >= n) return;           // wave-uniform: EXEC stays all-ones for WMMA
  const int col0 = nTile * 64;
  const int l16  = lane & 15;
  const int koff = (lane >> 4) << 1;          // lanes 0-15: K+0,1 ; 16-31: K+2,3
  const float* xrow = X + (size_t)(row0 + l16) * K + koff;
  v8f acc0 = {}, acc1 = {}, acc2 = {}, acc3 = {};
  for (int k = 0; k < K; k += 4) {
    v2f a = *(const v2f*)(xrow + k);          // A: 16x4 f32 fragment
    const float* w0 = Wm + (size_t)(k + koff) * 256 + col0 + l16;
    const float* w1 = w0 + 256;
    v2f b0 = { w0[0],  w1[0]  };              // B: 4x16 f32 fragments
    v2f b1 = { w0[16], w1[16] };
    v2f b2 = { w0[32], w1[32] };
    v2f b3 = { w0[48], w1[48] };
    acc0 = __builtin_amdgcn_wmma_f32_16x16x4_f32(false, a, false, b0, (short)0, acc0, false, false);
    acc1 = __builtin_amdgcn_wmma_f32_16x16x4_f32(false, a, false, b1, (short)0, acc1, false, false);
    acc2 = __builtin_amdgcn_wmma_f32_16x16x4_f32(false, a, false, b2, (short)0, acc2, false, false);
    acc3 = __builtin_amdgcn_wmma_f32_16x16x4_f32(false, a, false, b3, (short)0, acc3, false, false);
  }
  // D layout: VGPR i -> row row0 + 8*(lane>=16) + i, col col0 + lane%16
  const int r0 = row0 + ((lane >> 4) << 3);
  float* out = C + (size_t)r0 * 256 + col0 + l16;
#pragma unroll
  for (int i = 0; i < 8; ++i) {
    out[(size_t)i * 256 +  0] = acc0[i];
    out[(size_t)i * 256 + 16] = acc1[i];
    out[(size_t)i * 256 + 32] = acc2[i];
    out[(size_t)i * 256 + 48] = acc3[i];
  }
}

// ---------------- per-(node,head) attention dots: es/ed ---------------------
__global__ __launch_bounds__(256) void attn_scores(
    const float* __restrict__ h, const float* __restrict__ a_src,
    const float* __restrict__ a_dst, float* __restrict__ es,
    float* __restrict__ ed, int n, int H, int Cc) {
  int wid  = (int)((blockIdx.x * blockDim.x + threadIdx.x) >> 5);
  int lane = (int)(threadIdx.x & 31);
  if (wid >= n * H) return;
  int node = wid / H, head = wid - node * H;
  const float* hp  = h + (size_t)node * 256 + head * Cc;
  const float* pas = a_src + head * Cc;
  const float* pad = a_dst + head * Cc;
  float ss = 0.f, dd = 0.f;
  for (int c = lane; c < Cc; c += 32) {
    float hv = hp[c];
    ss += hv * pas[c];
    dd += hv * pad[c];
  }
#pragma unroll
  for (int off = 16; off > 0; off >>= 1) {
    ss += __shfl_down(ss, off);
    dd += __shfl_down(dd, off);
  }
  if (lane == 0) { es[wid] = ss; ed[wid] = dd; }
}

// ---------------- init kernels ----------------------------------------------
__global__ __launch_bounds__(256) void init_node(unsigned* __restrict__ mmax,
                                                 float* __restrict__ ssum, int cnt) {
  int t = blockIdx.x * blockDim.x + threadIdx.x;
  if (t >= cnt) return;
  mmax[t] = 0u;       // decodes below any finite float
  ssum[t] = 0.f;
}

__global__ __launch_bounds__(256) void init_out(float* __restrict__ out,
                                                const float* __restrict__ bias,
                                                long long cnt) {
  long long t = (long long)blockIdx.x * blockDim.x + threadIdx.x;
  if (t >= cnt) return;
  out[t] = bias[(int)(t & 255)];
}

// ---------------- edge pass 1: leaky-relu scores + segment max --------------
__global__ __launch_bounds__(256) void edge_max(
    const int* __restrict__ ei, const float* __restrict__ es,
    const float* __restrict__ ed, float* __restrict__ evals,
    unsigned* __restrict__ mmax, int E, int n, int H) {
  long long t = (long long)blockIdx.x * blockDim.x + threadIdx.x;
  long long tot = (long long)(E + n) * H;
  if (t >= tot) return;
  int e = (int)(t / H), hd = (int)(t - (long long)e * H);
  int s, d;
  if (e < E) { s = ei[e]; d = ei[E + e]; } else { s = e - E; d = s; }
  float v = es[(size_t)s * H + hd] + ed[(size_t)d * H + hd];
  v = v > 0.f ? v : LRELU_SLOPE * v;
  evals[t] = v;
  atomicMax(mmax + (size_t)d * H + hd, fenc(v));
}

// ---------------- edge pass 2: exp(e - m) + segment sum ---------------------
__global__ __launch_bounds__(256) void edge_expsum(
    const int* __restrict__ ei, float* __restrict__ evals,
    const unsigned* __restrict__ mmax, float* __restrict__ ssum,
    int E, int n, int H) {
  long long t = (long long)blockIdx.x * blockDim.x + threadIdx.x;
  long long tot = (long long)(E + n) * H;
  if (t >= tot) return;
  int e = (int)(t / H), hd = (int)(t - (long long)e * H);
  int d = (e < E) ? ei[E + e] : (e - E);
  float m  = fdec(mmax[(size_t)d * H + hd]);
  float ex = expf(evals[t] - m);
  evals[t] = ex;
  atomicAdd(ssum + (size_t)d * H + hd, ex);
}

// ---------------- edge pass 3: out[dst] += h[src] * alpha -------------------
// 64 threads per edge, float4 per thread (256 channels).
__global__ __launch_bounds__(256) void edge_aggregate(
    const int* __restrict__ ei, const float* __restrict__ evals,
    const float* __restrict__ ssum, const float* __restrict__ h,
    float* __restrict__ out, int E, int n, int H) {
  long long t = (long long)blockIdx.x * blockDim.x + threadIdx.x;
  int e   = (int)(t >> 6);
  int sub = (int)(t & 63);
  if (e >= E + n) return;
  int s, d;
  if (e < E) { s = ei[e]; d = ei[E + e]; } else { s = e - E; d = s; }
  int c0   = sub << 2;
  int head = (c0 * H) >> 8;                     // c0 / (256/H)
  float alpha = evals[(size_t)e * H + head] /
                (ssum[(size_t)d * H + head] + 1e-16f);
  const float4 hv = *(const float4*)(h + (size_t)s * 256 + c0);
  float* op = out + (size_t)d * 256 + c0;
  atomicAdd(op + 0, hv.x * alpha);
  atomicAdd(op + 1, hv.y * alpha);
  atomicAdd(op + 2, hv.z * alpha);
  atomicAdd(op + 3, hv.w * alpha);
}

__global__ __launch_bounds__(256) void relu_k(float* __restrict__ x, long long cnt) {
  long long t = (long long)blockIdx.x * blockDim.x + threadIdx.x;
  if (t >= cnt) return;
  float v = x[t];
  x[t] = v > 0.f ? v : 0.f;
}

// ---------------- host side -------------------------------------------------
static inline int blocksFor(long long threads, int bs) {
  return (int)((threads + bs - 1) / bs);
}

static void run_layer(const float* xin, int K, const float* Wm, const float* as_,
                      const float* ad_, const float* b_, int H,
                      float* hbuf, float* outbuf,
                      float* es, float* ed, unsigned* mmax, float* ssum,
                      float* evals, const int* ei, int n, int E, bool do_relu,
                      hipStream_t stream) {
  const int Cc = 256 / H;
  const int Et = E + n;
  long long gemmThreads = ((long long)((n + 15) / 16) * 4) * 32;
  gemm_f32_wmma<<<blocksFor(gemmThreads, 256), 256, 0, stream>>>(xin, Wm, hbuf, n, K);
  attn_scores<<<blocksFor((long long)n * H * 32, 256), 256, 0, stream>>>(
      hbuf, as_, ad_, es, ed, n, H, Cc);
  init_node<<<blocksFor((long long)n * H, 256), 256, 0, stream>>>(mmax, ssum, n * H);
  edge_max<<<blocksFor((long long)Et * H, 256), 256, 0, stream>>>(
      ei, es, ed, evals, mmax, E, n, H);
  edge_expsum<<<blocksFor((long long)Et * H, 256), 256, 0, stream>>>(
      ei, evals, mmax, ssum, E, n, H);
  init_out<<<blocksFor((long long)n * 256, 256), 256, 0, stream>>>(
      outbuf, b_, (long long)n * 256);
  edge_aggregate<<<blocksFor((long long)Et * 64, 256), 256, 0, stream>>>(
      ei, evals, ssum, hbuf, outbuf, E, n, H);
  if (do_relu)
    relu_k<<<blocksFor((long long)n * 256, 256), 256, 0, stream>>>(
        outbuf, (long long)n * 256);
}

extern "C" void kernel_launch(void* const* d_in, const int* in_sizes, int n_in,
                              void* d_out, int out_size, void* d_ws, size_t ws_size,
                              hipStream_t stream) {
  const float* x   = (const float*)d_in[0];
  const int*   ei  = (const int*)d_in[1];
  const float* W0  = (const float*)d_in[2];
  const float* as0 = (const float*)d_in[3];
  const float* ad0 = (const float*)d_in[4];
  const float* b0  = (const float*)d_in[5];
  const float* W1  = (const float*)d_in[6];
  const float* as1 = (const float*)d_in[7];
  const float* ad1 = (const float*)d_in[8];
  const float* b1  = (const float*)d_in[9];
  const float* W2  = (const float*)d_in[10];
  const float* as2 = (const float*)d_in[11];
  const float* ad2 = (const float*)d_in[12];
  const float* b2  = (const float*)d_in[13];

  const int n = in_sizes[0] / 384;   // 50000
  const int E = in_sizes[1] / 2;     // 400000
  const size_t nh = (size_t)n * 256;

  // workspace layout (fp32 elements)
  float*    A     = (float*)d_ws;                 // GEMM output h       [n*256]
  float*    B     = A + nh;                       // aggregated output   [n*256]
  float*    es    = B + nh;                       // [n*4]
  float*    ed    = es + (size_t)n * 4;           // [n*4]
  unsigned* mmax  = (unsigned*)(ed + (size_t)n * 4); // [n*4]
  float*    ssum  = (float*)(mmax + (size_t)n * 4);  // [n*4]
  float*    evals = ssum + (size_t)n * 4;         // [(E+n)*4]

  // layer 0: x(384) -> A -> B, relu
  run_layer(x, 384, W0, as0, ad0, b0, 4, A, B, es, ed, mmax, ssum, evals,
            ei, n, E, true, stream);
  // layer 1: B(256) -> A -> B, relu
  run_layer(B, 256, W1, as1, ad1, b1, 4, A, B, es, ed, mmax, ssum, evals,
            ei, n, E, true, stream);
  // layer 2: B(256) -> A -> d_out, single head, no relu
  run_layer(B, 256, W2, as2, ad2, b2, 1, A, (float*)d_out, es, ed, mmax, ssum,
            evals, ei, n, E, false, stream);
}